// SquareSensor_71786083385668
// MI455X (gfx1250) — compile-verified
//
#include <hip/hip_runtime.h>
#include <cstdint>

// ---- Sensor static config (mirrors reference) ----
#define SENSOR_W 1024
#define SENSOR_H 1024
static constexpr float kX0 = -3.0f;
static constexpr float kY0 = -3.0f;
static constexpr float kDX = 6.0f / 1024.0f;   // exactly representable (3*2^-9)
static constexpr float kDY = 6.0f / 1024.0f;

// ---- Tiling ----
static constexpr int TILE            = 128;               // hits per wave-tile
static constexpr int WAVES_PER_BLOCK = 8;                 // 256 threads, wave32
static constexpr int BUF_FLOATS      = 3 * TILE;          // x|y|v = 384 floats
static constexpr int WAVE_FLOATS     = 2 * BUF_FLOATS;    // double buffered = 768

// ---------------- CDNA5 inline-asm helpers ----------------

// Async DMA: 16 bytes/lane from global into LDS; tracked by ASYNCcnt.
__device__ __forceinline__ void async_b128(uint32_t lds_byte, const float* gaddr) {
  asm volatile("global_load_async_to_lds_b128 %0, %1, off th:TH_LOAD_NT"
               :: "v"(lds_byte), "v"(gaddr) : "memory");
}

// Async loads complete in order: with 6 in flight, cnt<=3 => first tile landed.
__device__ __forceinline__ void wait_async_le3() {
  asm volatile("s_wait_asynccnt 0x3" ::: "memory");
}
__device__ __forceinline__ void wait_async_0() {
  asm volatile("s_wait_asynccnt 0x0" ::: "memory");
}

// Non-returning fp32 atomic add at device scope (L2 atomic units; STOREcnt).
__device__ __forceinline__ void atomic_add_f32_dev(float* base, uint32_t byte_off,
                                                   float val) {
  asm volatile("global_atomic_add_f32 %0, %1, %2 scope:SCOPE_DEV"
               :: "v"(byte_off), "v"(val), "s"(base) : "memory");
}

// ---------------- binning ----------------

__device__ __forceinline__ void bin_one(float xx, float yy, float vv,
                                        float* __restrict__ out) {
  // Match reference semantics: floor((p - p0) / dp), then bounds check.
  const float fx = floorf((xx - kX0) / kDX);
  const float fy = floorf((yy - kY0) / kDY);
  const int xi = (int)fx;
  const int yi = (int)fy;
  if (xi >= 0 && xi < SENSOR_W && yi >= 0 && yi < SENSOR_H) {
    const uint32_t byte_off = ((uint32_t)(yi * SENSOR_W + xi)) << 2;
    atomic_add_f32_dev(out, byte_off, vv);
  }
}

// ---------------- kernels ----------------

__global__ __launch_bounds__(256) void zero_image(float4* __restrict__ out, int n4) {
  const int stride = (int)(gridDim.x * blockDim.x);
  const float4 z = make_float4(0.f, 0.f, 0.f, 0.f);
  for (int i = (int)(blockIdx.x * blockDim.x + threadIdx.x); i < n4; i += stride)
    out[i] = z;
}

__global__ __launch_bounds__(256) void sensor_hist(const float* __restrict__ xs,
                                                   const float* __restrict__ ys,
                                                   const float* __restrict__ vs,
                                                   float* __restrict__ out, int n) {
  __shared__ float smem[WAVES_PER_BLOCK * WAVE_FLOATS];  // 24 KB / block

  const int lane  = (int)(threadIdx.x & 31);
  const int wslot = (int)(threadIdx.x >> 5);
  const int gwave = (int)((blockIdx.x * blockDim.x + threadIdx.x) >> 5);
  const int nwave = (int)((gridDim.x * blockDim.x) >> 5);
  const int ntiles = n / TILE;

  float* wptr = &smem[wslot * WAVE_FLOATS];
  // Generic LDS pointer low 32 bits == LDS byte offset (aperture holds [63:32]).
  const uint32_t wbyte = (uint32_t)(uintptr_t)(void*)wptr;

  auto issue = [&](int tile, int buf) {
    const uint32_t b = wbyte + (uint32_t)buf * (uint32_t)(BUF_FLOATS * 4);
    const int g = tile * TILE + lane * 4;                 // 4 floats / lane
    async_b128(b + (uint32_t)(lane * 16),        xs + g); // x -> [0,512)
    async_b128(b + 512u + (uint32_t)(lane * 16), ys + g); // y -> [512,1024)
    async_b128(b + 1024u + (uint32_t)(lane * 16), vs + g);// v -> [1024,1536)
  };

  // Double-buffered pipeline: prefetch tile (cur+nwave) while binning tile cur.
  int cur = gwave;
  int buf = 0;
  if (cur < ntiles) issue(cur, 0);
  while (cur < ntiles) {
    const int nxt = cur + nwave;
    if (nxt < ntiles) {
      issue(nxt, buf ^ 1);
      wait_async_le3();       // current tile's 3 loads are done (in-order)
    } else {
      wait_async_0();
    }

    const float* bp = wptr + buf * BUF_FLOATS;
    const float4 x4 = ((const float4*)(bp))[lane];
    const float4 y4 = ((const float4*)(bp + TILE))[lane];
    const float4 v4 = ((const float4*)(bp + 2 * TILE))[lane];

    bin_one(x4.x, y4.x, v4.x, out);
    bin_one(x4.y, y4.y, v4.y, out);
    bin_one(x4.z, y4.z, v4.z, out);
    bin_one(x4.w, y4.w, v4.w, out);

    buf ^= 1;
    cur = nxt;
  }

  // Tail (n not a multiple of TILE): plain loads, same binning.
  const int gtid = (int)(blockIdx.x * blockDim.x + threadIdx.x);
  const int nthreads = (int)(gridDim.x * blockDim.x);
  for (int i = ntiles * TILE + gtid; i < n; i += nthreads)
    bin_one(xs[i], ys[i], vs[i], out);
}

// ---------------- launch ----------------

extern "C" void kernel_launch(void* const* d_in, const int* in_sizes, int n_in,
                              void* d_out, int out_size, void* d_ws, size_t ws_size,
                              hipStream_t stream) {
  const float* x = (const float*)d_in[0];
  const float* y = (const float*)d_in[1];
  const float* v = (const float*)d_in[2];
  float* out = (float*)d_out;
  const int n  = in_sizes[0];          // 16,777,216
  const int n4 = out_size / 4;         // 1024*1024/4 float4s

  zero_image<<<1024, 256, 0, stream>>>((float4*)out, n4);
  // 2048 blocks * 8 waves = 16384 waves; 131072 tiles -> 8 tiles/wave.
  sensor_hist<<<2048, 256, 0, stream>>>(x, y, v, out, n);
}